// Net_7370163880040
// MI455X (gfx1250) — compile-verified
//
#include <hip/hip_runtime.h>
#include <math.h>

typedef __attribute__((ext_vector_type(2))) float v2f;
typedef __attribute__((ext_vector_type(8))) float v8f;
typedef __attribute__((ext_vector_type(4))) unsigned int v4u;
typedef __attribute__((ext_vector_type(4))) int v4i;
typedef __attribute__((ext_vector_type(8))) int v8i;

#define NN 20000
#define NP 20096   // NN rounded up to multiple of 128 (node-tile padding)
#define NE 120000
#define MM 25
#define SS 8

#if defined(__has_builtin)
#if __has_builtin(__builtin_amdgcn_tensor_load_to_lds) && \
    __has_builtin(__builtin_amdgcn_s_wait_tensorcnt)
#define USE_TDM 1
#endif
#endif
#ifndef USE_TDM
#define USE_TDM 0
#endif

// ---------------------------------------------------------------- utilities
__global__ void zero_kernel(float* __restrict__ p, int n) {
  int i = blockIdx.x * 256 + threadIdx.x;
  if (i < n) p[i] = 0.f;
}

// copy x into padded h buffer, zero-filling the pad rows
__global__ void copyx_kernel(const float* __restrict__ x, float* __restrict__ h,
                             int n_valid, int n_tot) {
  int i = blockIdx.x * 256 + threadIdx.x;
  if (i < n_tot) h[i] = (i < n_valid) ? x[i] : 0.f;
}

// rdeg = 1 / max(deg, 1); deg is zero-padded so pad -> 1
__global__ void rdeg_kernel(const float* __restrict__ deg,
                            float* __restrict__ rdeg, int np) {
  int i = blockIdx.x * 256 + threadIdx.x;
  if (i < np) {
    float d = deg[i];
    rdeg[i] = 1.f / (d < 1.f ? 1.f : d);
  }
}

// Spline basis (degree-1 open B-spline), cell indices, degree histogram.
__global__ void basis_deg_kernel(const float* __restrict__ pseudo,
                                 const int* __restrict__ dst,
                                 float* __restrict__ basis,
                                 int* __restrict__ wi,
                                 float* __restrict__ deg, int E) {
  int e = blockIdx.x * 256 + threadIdx.x;
  if (e >= E) return;
  float fr[3]; int fl[3];
#pragma unroll
  for (int d = 0; d < 3; ++d) {
    float v = pseudo[e * 3 + d] * (float)(MM - 1);
    float f = floorf(v);
    fr[d] = v - f;
    fl[d] = (int)f;
  }
#pragma unroll
  for (int s = 0; s < SS; ++s) {
    float b = 1.f;
    int w = 0, stride = 1;
#pragma unroll
    for (int d = 0; d < 3; ++d) {
      int bit = (s >> d) & 1;
      b *= bit ? fr[d] : (1.f - fr[d]);
      int id = fl[d] + bit;
      if (id >= MM) id -= MM;
      w += id * stride;
      stride *= MM;
    }
    basis[e * SS + s] = b;
    wi[e * SS + s] = w;
  }
  atomicAdd(&deg[dst[e]], 1.f);
}

// ---------------------------------------------------------------- edge kernel
// One wave per edge, lane = output channel; 8 edges per block.
// Per-s dot (two interleaved accumulators) then one fma(basis, dot, acc),
// matching the reference's  msg += basis[s] * (xj @ Ws)  order.
// Weight rows are coalesced 128B reads; next spline cell row is prefetched.
template <int FI, int FO>
__global__ void edge_kernel(const float* __restrict__ h,
                            const float* __restrict__ W,
                            const int* __restrict__ src,
                            const int* __restrict__ dst,
                            const float* __restrict__ basis,
                            const int* __restrict__ wi,
                            float* __restrict__ agg) {
  __shared__ float sb[8][SS];
  __shared__ int   swi[8][SS];
  __shared__ float sxj[8][FI];

  int o = threadIdx.x;   // 0..31
  int ey = threadIdx.y;  // 0..7
  int e = blockIdx.x * 8 + ey;

  if (o < SS) sb[ey][o] = basis[e * SS + o];
  else if (o < 2 * SS) swi[ey][o - SS] = wi[e * SS + (o - SS)];
  int sn = src[e];
  if (o < FI) sxj[ey][o] = h[sn * FI + o];
  __syncthreads();

  if (o < FO) {
    float acc = 0.f;
#pragma unroll
    for (int s = 0; s < SS; ++s) {
      const float* Wp = W + (size_t)swi[ey][s] * (FI * FO);
      if (s + 1 < SS)
        __builtin_prefetch(W + (size_t)swi[ey][s + 1] * (FI * FO) + o, 0, 0);
      float d0 = 0.f, d1 = 0.f;
#pragma unroll
      for (int f = 0; f < FI; f += 2) {
        d0 = fmaf(sxj[ey][f], Wp[f * FO + o], d0);
        d1 = fmaf(sxj[ey][f + 1], Wp[(f + 1) * FO + o], d1);
      }
      acc = fmaf(sb[ey][s], d0 + d1, acc);
    }
    atomicAdd(&agg[(size_t)dst[e] * FO + o], acc);
  }
}

// ---------------------------------------------------------------- node kernel
// out = agg*rdeg + h@root + bias, ELU, BN stats.
// h@root with V_WMMA_F32_16X16X4_F32; every fragment comes from zero-padded
// LDS, so fragment loads are unconditional ds_loads. The 128-node h tile is
// DMA'd into LDS with the Tensor Data Mover (tensor_load_to_lds +
// s_wait_tensorcnt) when available. agg/rdeg tiles are LDS-staged too, so
// the epilogue touches global memory only for the final stores.
template <int FI, int FO, bool LAST>
__global__ void node_kernel(const float* __restrict__ h,     // padded NP*FI
                            const float* __restrict__ agg,   // padded, zeroed
                            const float* __restrict__ rdeg,  // padded
                            const float* __restrict__ root,
                            const float* __restrict__ bias,
                            float* __restrict__ hout,        // padded unless LAST
                            float* __restrict__ stats, int n) {
  constexpr int NT = (FO + 15) / 16;
  __shared__ float s_root[32 * 32];        // zero-padded
  __shared__ float s_bias[32];             // zero-padded
  __shared__ float s_h[128 * FI + 8];      // 128-node input tile (+slack)
  __shared__ float s_agg[128 * FO + 32];   // 128-node agg tile (+slack)
  __shared__ float s_rdeg[128];
  __shared__ float s_sum[32], s_sq[32];

  int tid = threadIdx.x;
  for (int i = tid; i < 32 * 32; i += 256) {
    int k = i >> 5, col = i & 31;
    s_root[i] = (k < FI && col < FO) ? root[k * FO + col] : 0.f;
  }
  if (tid < 32) {
    s_bias[tid] = (tid < FO) ? bias[tid] : 0.f;
    s_sum[tid] = 0.f;
    s_sq[tid] = 0.f;
  }
  int base = blockIdx.x * (128 * FI);
#if USE_TDM
  if (tid < 32) {  // one wave issues the TDM transfer (EXEC is ignored)
    unsigned long long gaddr = (unsigned long long)(uintptr_t)(h + base);
    unsigned lds_addr = (unsigned)(uintptr_t)(&s_h[0]);  // low 32b = LDS offset
    const unsigned nelem = 128u * FI;  // one contiguous row of f32
    v4u g0;
    g0.x = 1u;                                     // count=1, load, user
    g0.y = lds_addr;                               // lds_addr
    g0.z = (unsigned)(gaddr & 0xffffffffull);      // global_addr[31:0]
    g0.w = (unsigned)((gaddr >> 32) & 0x01ffffffull) | 0x80000000u;  // type=2
    v8i g1;
    g1[0] = 0x00020000;                            // data_size = 4B
    g1[1] = (int)((nelem & 0xffffu) << 16);        // tensor_dim0 lo16
    g1[2] = (int)((nelem >> 16) | (1u << 16));     // dim0 hi16 | tensor_dim1=1
    g1[3] = (int)(nelem << 16);                    // tile_dim0 = nelem
    g1[4] = 1;                                     // tile_dim1=1, tile_dim2=0
    g1[5] = (int)nelem;                            // tensor_dim0_stride lo32
    g1[6] = 0;
    g1[7] = 0;
    v4i gz = {0, 0, 0, 0};
#if __clang_major__ >= 23
    v8i gz8 = {0, 0, 0, 0, 0, 0, 0, 0};
    __builtin_amdgcn_tensor_load_to_lds(g0, g1, gz, gz, gz8, 0);
#else
    __builtin_amdgcn_tensor_load_to_lds(g0, g1, gz, gz, 0);
#endif
    __builtin_amdgcn_s_wait_tensorcnt(0);
  }
#else
  for (int i = tid; i < 128 * FI; i += 256) s_h[i] = h[base + i];
#endif
  {  // stage agg tile + rdeg tile (coalesced)
    int abase = blockIdx.x * (128 * FO);
    for (int i = tid; i < 128 * FO; i += 256) s_agg[i] = agg[abase + i];
    if (tid < 128) s_rdeg[tid] = rdeg[blockIdx.x * 128 + tid];
  }
  __syncthreads();

  int lane = tid & 31;
  int wv = tid >> 5;
  int arow = lane & 15;
  int kpair = (lane >> 4) * 2;
  int lnode = wv * 16 + arow;
  int node_base = blockIdx.x * 128 + wv * 16;

  v8f c[NT] = {};
#pragma unroll
  for (int k0 = 0; k0 < FI; k0 += 4) {
    int ka0 = k0 + kpair;
    v2f a;
    {  // guard folds away unless FI % 4 != 0 (only FI==2)
      float t0 = s_h[lnode * FI + ka0];
      float t1 = s_h[lnode * FI + ka0 + 1];
      a.x = (ka0 < FI) ? t0 : 0.f;
      a.y = (ka0 + 1 < FI) ? t1 : 0.f;
    }
#pragma unroll
    for (int t = 0; t < NT; ++t) {
      int col = t * 16 + (lane & 15);
      v2f b;
      b.x = s_root[ka0 * 32 + col];        // zero-padded rows -> safe
      b.y = s_root[(ka0 + 1) * 32 + col];
      c[t] = __builtin_amdgcn_wmma_f32_16x16x4_f32(
          false, a, false, b, (short)0, c[t], false, false);
    }
  }

#pragma unroll
  for (int t = 0; t < NT; ++t) {
    int col = t * 16 + (lane & 15);
    float ls = 0.f, lq = 0.f;
#pragma unroll
    for (int j = 0; j < 8; ++j) {
      int lrow = wv * 16 + (lane >> 4) * 8 + j;
      int node = node_base + (lane >> 4) * 8 + j;
      float val = c[t][j] + s_agg[lrow * FO + col] * s_rdeg[lrow] + s_bias[col];
      val = val > 0.f ? val : (__expf(val) - 1.f);  // ELU
      if (LAST) {
        if (node < n && col < FO) hout[(size_t)node * FO + col] = val;
      } else {
        if (FO >= 16 || col < FO)                   // hout is padded
          hout[(size_t)node * FO + col] = val;
        float sv = (node < n) ? val : 0.f;
        ls += sv;
        lq += sv * sv;
      }
    }
    if (!LAST) {
      atomicAdd(&s_sum[col], ls);
      atomicAdd(&s_sq[col], lq);
    }
  }
  if (!LAST) {
    __syncthreads();
    if (tid < FO) {
      atomicAdd(&stats[tid], s_sum[tid]);
      atomicAdd(&stats[32 + tid], s_sq[tid]);
    }
  }
}

// ---------------------------------------------------------------- batch norm
template <int FO>
__global__ void bn_kernel(float* __restrict__ h, const float* __restrict__ stats,
                          const float* __restrict__ gamma,
                          const float* __restrict__ beta, int n) {
  int i = blockIdx.x * 256 + threadIdx.x;
  if (i >= NP * FO) return;
  int col = i % FO;
  float inv_n = 1.f / (float)n;
  float mu = stats[col] * inv_n;
  float var = stats[32 + col] * inv_n - mu * mu;
  h[i] = gamma[col] * (h[i] - mu) * rsqrtf(var + 1e-5f) + beta[col];
}

// ---------------------------------------------------------------- host side
template <int FI, int FO, bool LAST>
static void run_layer_t(const float* hin, const float* W, const float* root,
                        const float* bias, const float* gamma, const float* beta,
                        float* hout, const int* src, const int* dst,
                        const float* basis, const int* wi, float* agg,
                        float* rdeg, float* stats, hipStream_t stream) {
  dim3 eb(32, 8);
  edge_kernel<FI, FO><<<NE / 8, eb, 0, stream>>>(hin, W, src, dst, basis, wi,
                                                 agg);
  node_kernel<FI, FO, LAST><<<NP / 128, 256, 0, stream>>>(
      hin, agg, rdeg, root, bias, hout, stats, NN);
  if (!LAST)
    bn_kernel<FO><<<(NP * FO + 255) / 256, 256, 0, stream>>>(hout, stats, gamma,
                                                             beta, NN);
}

extern "C" void kernel_launch(void* const* d_in, const int* in_sizes, int n_in,
                              void* d_out, int out_size, void* d_ws,
                              size_t ws_size, hipStream_t stream) {
  (void)in_sizes; (void)n_in; (void)out_size; (void)ws_size;
  const float* x = (const float*)d_in[0];
  const int* ei = (const int*)d_in[1];
  const float* pseudo = (const float*)d_in[2];
  const int* src = ei;
  const int* dst = ei + NE;

  float* ws = (float*)d_ws;
  float* basis = ws;                     //   960000 f
  int* wi = (int*)(ws + 960000);         //   960000 i
  float* deg = ws + 1920000;             //    20096 f
  float* rdeg = ws + 1940096;            //    20096 f
  float* hA = ws + 1960192;              //   643072 f (NP*32)
  float* hB = ws + 2603264;              //   643072 f
  float* agg = ws + 3246336;             //   643072 f
  float* stats = ws + 3889408;           //       64 f (contiguous with agg)

  zero_kernel<<<(NP + 255) / 256, 256, 0, stream>>>(deg, NP);
  copyx_kernel<<<(NP * 2 + 255) / 256, 256, 0, stream>>>(x, hA, NN * 2, NP * 2);
  basis_deg_kernel<<<(NE + 255) / 256, 256, 0, stream>>>(pseudo, dst, basis, wi,
                                                         deg, NE);
  rdeg_kernel<<<(NP + 255) / 256, 256, 0, stream>>>(deg, rdeg, NP);

  const float* hin = hA;
  for (int layer = 0; layer < 12; ++layer) {
    const float* W = (const float*)d_in[3 + 5 * layer + 0];
    const float* root = (const float*)d_in[3 + 5 * layer + 1];
    const float* bias = (const float*)d_in[3 + 5 * layer + 2];
    const float* gamma = (const float*)d_in[3 + 5 * layer + 3];
    const float* beta = (const float*)d_in[3 + 5 * layer + 4];
    float* hout = (layer == 11) ? (float*)d_out : ((layer & 1) ? hA : hB);

    // zero agg (NP*32) + stats (64) in one pass
    zero_kernel<<<(643136 + 255) / 256, 256, 0, stream>>>(agg, 643136);

    switch (layer) {
      case 0:  run_layer_t<2, 8, false>(hin, W, root, bias, gamma, beta, hout, src, dst, basis, wi, agg, rdeg, stats, stream); break;
      case 1:  run_layer_t<8, 16, false>(hin, W, root, bias, gamma, beta, hout, src, dst, basis, wi, agg, rdeg, stats, stream); break;
      case 2:  run_layer_t<16, 32, false>(hin, W, root, bias, gamma, beta, hout, src, dst, basis, wi, agg, rdeg, stats, stream); break;
      case 3: case 4: case 5: case 6: case 7: case 8:
               run_layer_t<32, 32, false>(hin, W, root, bias, gamma, beta, hout, src, dst, basis, wi, agg, rdeg, stats, stream); break;
      case 9:  run_layer_t<32, 16, false>(hin, W, root, bias, gamma, beta, hout, src, dst, basis, wi, agg, rdeg, stats, stream); break;
      case 10: run_layer_t<16, 8, false>(hin, W, root, bias, gamma, beta, hout, src, dst, basis, wi, agg, rdeg, stats, stream); break;
      case 11: run_layer_t<8, 1, true>(hin, W, root, bias, gamma, beta, hout, src, dst, basis, wi, agg, rdeg, stats, stream); break;
    }
    hin = hout;
  }
}